// PendulumSystem_55886114455898
// MI455X (gfx1250) — compile-verified
//
#include <hip/hip_runtime.h>
#include <math.h>
#include <stdint.h>

// Pendulum + ANFIS scan: 131072 independent cases x 100 sequential steps.
// One thread per case, state in registers. Output (N,3,100) staged in LDS in
// 20-step tiles, drained with CDNA5 async LDS->global b128 stores (ASYNCcnt).
// Per-step transcendentals use hardware v_log_f32/v_exp_f32/sincos paths to
// shorten the serial dependence chain (the latency limiter; HBM floor ~6.8us).

#define BLOCK    256
#define T_TILE   20
#define N_TILES  5            // 5 * 20 = 100 steps
#define PER_CASE 60           // 3 outputs * 20 steps per tile (floats)

// pow(x, b) for x >= 0 via hardware log2/exp2 (v_log_f32 / v_exp_f32).
// x==0: log2->-inf, b*-inf=-inf (b>0), exp2->0  == 0**b. Correct here.
__device__ __forceinline__ float pow_nn(float x, float b) {
    return __builtin_amdgcn_exp2f(b * __builtin_amdgcn_logf(x));
}

__global__ __launch_bounds__(BLOCK)
void pendulum_anfis_kernel(const float* __restrict__ x,
                           const float* __restrict__ mf_a,
                           const float* __restrict__ mf_b,
                           const float* __restrict__ mf_c,
                           const float* __restrict__ coeff,
                           float* __restrict__ out,
                           int N)
{
    __shared__ float lds[BLOCK * PER_CASE];   // 60 KB, 16B-aligned per-case rows

    const int tid = threadIdx.x;
    const int n0  = blockIdx.x * BLOCK;
    const int n   = n0 + tid;
    const bool active = (n < N);

    const float DEG2RAD = 0.017453292519943295f;
    const float INV_TOT = 1.0f / 1.1f;        // 1/(M_C+M_P)

    // Uniform parameters (layouts: mf_*[dim*2+mf], coeff[r*3+k] for shape (4,1,3))
    const float ia00 = 1.0f / mf_a[0], ia01 = 1.0f / mf_a[1];
    const float ia10 = 1.0f / mf_a[2], ia11 = 1.0f / mf_a[3];
    const float b00 = mf_b[0], b01 = mf_b[1], b10 = mf_b[2], b11 = mf_b[3];
    const float c00 = mf_c[0], c01 = mf_c[1], c10 = mf_c[2], c11 = mf_c[3];
    float k[12];
    #pragma unroll
    for (int i = 0; i < 12; ++i) k[i] = coeff[i];

    float th = 0.0f, dth = 0.0f;
    if (active) { th = x[2 * n]; dth = x[2 * n + 1]; }

    float* const myrow = &lds[tid * PER_CASE];

    for (int tile = 0; tile < N_TILES; ++tile) {
        // ---- integrate 20 steps, stage into LDS ----
        #pragma unroll 2
        for (int tt = 0; tt < T_TILE; ++tt) {
            // ANFIS membership: mf = 1 / (1 + (d^2)^b)
            float d;
            d = (th  - c00) * ia00; float m00 = 1.0f / (1.0f + pow_nn(d * d, b00));
            d = (th  - c01) * ia01; float m01 = 1.0f / (1.0f + pow_nn(d * d, b01));
            d = (dth - c10) * ia10; float m10 = 1.0f / (1.0f + pow_nn(d * d, b10));
            d = (dth - c11) * ia11; float m11 = 1.0f / (1.0f + pow_nn(d * d, b11));

            float w0 = m00 * m10, w1 = m00 * m11, w2 = m01 * m10, w3 = m01 * m11;
            float wsum = fmaxf(w0 + w1 + w2 + w3, 1e-12f);

            float t0 = k[0] * th + k[1] * dth + k[2];
            float t1 = k[3] * th + k[4] * dth + k[5];
            float t2 = k[6] * th + k[7] * dth + k[8];
            float t3 = k[9] * th + k[10] * dth + k[11];
            float f  = (w0 * t0 + w1 * t1 + w2 * t2 + w3 * t3) / wsum;

            // pendulum dynamics (single shared range reduction for sin/cos)
            float rth  = th * DEG2RAD;
            float rdth = dth * DEG2RAD;
            float s, c;
            sincosf(rth, &s, &c);
            float numer = 9.81f * s + c * ((-f - 0.05f * rdth * rdth * s) * INV_TOT);
            float denom = 0.5f * (4.0f / 3.0f - 0.1f * c * c * INV_TOT);
            float ddth  = (numer / denom) / DEG2RAD;

            th  = th  + 0.01f * dth;   // uses old dtheta
            dth = dth + 0.01f * ddth;

            myrow[tt]      = th;       // o = 0
            myrow[20 + tt] = dth;      // o = 1
            myrow[40 + tt] = f;        // o = 2
        }

        __syncthreads();               // make all staged LDS data visible

        // ---- cooperative async drain: 3840 x b128 (16B) per tile ----
        const int tb = tile * T_TILE;
        for (int u = tid; u < BLOCK * 15; u += BLOCK) {
            const int cse = u / 15;
            const int seg = u - cse * 15;        // 0..14
            const int o   = seg / 5;             // output row
            const int q   = seg - o * 5;         // quad within row
            const int gn  = n0 + cse;
            if (gn < N) {
                const float* lp = &lds[cse * PER_CASE + seg * 4];
                unsigned loff = (unsigned)(uintptr_t)lp;            // low 32b = LDS byte addr
                float* gp = out + (size_t)gn * 300 + o * 100 + tb + q * 4;
                unsigned long long ga = (unsigned long long)(uintptr_t)gp;
                asm volatile("global_store_async_from_lds_b128 %0, %1, off"
                             :: "v"(ga), "v"(loff) : "memory");
            }
        }

        // ensure this wave's async DMA stores have consumed/committed LDS data,
        // then barrier so no other wave's DMA is still reading regions we reuse
        asm volatile("s_wait_asynccnt 0" ::: "memory");
        __syncthreads();
    }
}

extern "C" void kernel_launch(void* const* d_in, const int* in_sizes, int n_in,
                              void* d_out, int out_size, void* d_ws, size_t ws_size,
                              hipStream_t stream) {
    const float* x     = (const float*)d_in[0];
    const float* mf_a  = (const float*)d_in[1];
    const float* mf_b  = (const float*)d_in[2];
    const float* mf_c  = (const float*)d_in[3];
    const float* coeff = (const float*)d_in[4];
    float* out = (float*)d_out;

    const int N = in_sizes[0] / 2;               // (N,2) state
    const int grid = (N + BLOCK - 1) / BLOCK;
    pendulum_anfis_kernel<<<grid, BLOCK, 0, stream>>>(x, mf_a, mf_b, mf_c, coeff, out, N);
}